// Exponentiation_84679575208181
// MI455X (gfx1250) — compile-verified
//
#include <hip/hip_runtime.h>
#include <stdint.h>

#define HH 1024
#define WW 1024
#define NB 4
#define PLANE (HH * WW)  // 1048576 elements per (b,c) plane

typedef unsigned int v4u __attribute__((ext_vector_type(4)));
typedef int v8i __attribute__((ext_vector_type(8)));
typedef int v4i __attribute__((ext_vector_type(4)));

// phi0 = identity + u/128 ; identity[b,0,h,w]=h, identity[b,1,h,w]=w (analytic)
__global__ __launch_bounds__(256) void init_phi(const float* __restrict__ u,
                                                float* __restrict__ phi) {
    int idx = blockIdx.x * blockDim.x + threadIdx.x;  // float4 index
    int v = idx << 2;
    int w = v & (WW - 1);
    int h = (v >> 10) & (HH - 1);
    int c = (v >> 20) & 1;
    float4 uv = ((const float4*)u)[idx];
    float4 base;
    if (c == 0) {
        float f = (float)h;
        base = make_float4(f, f, f, f);
    } else {
        float f = (float)w;
        base = make_float4(f, f + 1.0f, f + 2.0f, f + 3.0f);
    }
    const float s = 1.0f / 128.0f;
    float4 r;
    r.x = base.x + uv.x * s;
    r.y = base.y + uv.y * s;
    r.z = base.z + uv.z * s;
    r.w = base.w + uv.w * s;
    ((float4*)phi)[idx] = r;
}

// One composition pass: pout[b,c,i,j] = bilerp(pin[b,c,:,:],
//      y = clamp(pin[b,1,j,i]), x = clamp(pin[b,0,j,i]))   (border clamp)
__global__ __launch_bounds__(256) void compose(const float* __restrict__ pin,
                                               float* __restrict__ pout) {
    // TDM lands the tile as [c][tj][ti] with a 1-dword pad after every 32
    // dwords (pad_interval=32, pad_amount=1) -> pitch 33, bank-conflict-free.
    __shared__ float s[2][32][33];

    const int tx = threadIdx.x;  // 0..31 (one wave32 per ty)
    const int ty = threadIdx.y;  // 0..7
    const int j0 = blockIdx.x << 5;
    const int i0 = blockIdx.y << 5;
    const int b  = blockIdx.z;

    const float* base0 = pin + (size_t)(b * 2) * PLANE;  // channel 0
    const float* base1 = base0 + PLANE;                  // channel 1

    // Warm the gather footprint (displacements are small: y~i, x~j).
    int tid = ty * 32 + tx;
    if (tid < 64) {
        int c = tid >> 5, r = tid & 31;
        __builtin_prefetch(pin + (size_t)(b * 2 + c) * PLANE +
                               (size_t)(i0 + r) * WW + j0, 0, 3);
    }

    // Phase 1: Tensor Data Mover pulls the transposed coordinate tile
    // (32 contiguous floats along i  x  32 rows along j  x  2 channel planes)
    // straight into LDS. TDM ignores EXEC and issues one DMA per executing
    // wave, so only wave ty==0 (uniform branch) issues it.
    if (ty == 0) {
        uint32_t lds_base = (uint32_t)(uintptr_t)(&s[0][0][0]);  // low 32 = LDS offset
        uint64_t gaddr = (uint64_t)(uintptr_t)(base0 + (size_t)j0 * WW + i0);

        // D# group 0: count=1 | lds_addr | global_addr[56:0] | type=2
        v4u g0 = { 1u,
                   lds_base,
                   (uint32_t)gaddr,
                   (uint32_t)((gaddr >> 32) & 0x1FFFFFFu) | (2u << 30) };

        // D# group 1:
        //  d0: data_size=2 (4B) | pad_enable | pad_interval=4 (32 dwords) | pad_amount=0 (1 dword)
        //  tensor_dim0=1024, tensor_dim1=1024, tile=32x32x2,
        //  dim0_stride=1024, dim1_stride=1048576 (channel plane)
        v8i g1 = { (int)0x01120000u,
                   (int)(1024u << 16),                 // tensor_dim0[15:0]
                   (int)(1024u << 16),                 // dim0[31:16]=0 | tensor_dim1[15:0]
                   (int)(32u << 16),                   // dim1[31:16]=0 | tile_dim0=32
                   (int)(32u | (2u << 16)),            // tile_dim1=32 | tile_dim2=2
                   (int)1024,                          // tensor_dim0_stride[31:0]
                   (int)0,                             // stride0[47:32]=0 | stride1[15:0]=0
                   (int)0x10 };                        // stride1[47:16] (1048576>>16)

        // D# group 2: tensor_dim2=2; group 3: unused
        v4i g2 = { 2, 0, 0, 0 };
        v4i g3 = { 0, 0, 0, 0 };

#if defined(__clang_major__) && (__clang_major__ >= 23)
        v8i g4 = { 0, 0, 0, 0, 0, 0, 0, 0 };
        __builtin_amdgcn_tensor_load_to_lds(g0, g1, g2, g3, g4, 0);
#else
        __builtin_amdgcn_tensor_load_to_lds(g0, g1, g2, g3, 0);
#endif
        __builtin_amdgcn_s_wait_tensorcnt(0);
    }
    __syncthreads();

    // Phase 2: compute with tx -> j so gathers (~diagonal) and stores coalesce.
    const float inv_b = 1.0f / 511.5f;
    for (int r = ty; r < 32; r += 8) {
        const int i = i0 + r;
        const int j = j0 + tx;

        float xf = s[0][tx][r];  // phi[b,0,j,i]
        float yf = s[1][tx][r];  // phi[b,1,j,i]

        // standardise -> unnormalize -> border clamp (mirrors reference math)
        float gx = (xf - 511.5f) * inv_b;
        float gy = (yf - 511.5f) * inv_b;
        float xs = (gx + 1.0f) * 0.5f * 1023.0f;
        float ys = (gy + 1.0f) * 0.5f * 1023.0f;
        xs = fminf(fmaxf(xs, 0.0f), 1023.0f);
        ys = fminf(fmaxf(ys, 0.0f), 1023.0f);

        float x0f = floorf(xs), y0f = floorf(ys);
        float wx = xs - x0f, wy = ys - y0f;
        int x0 = (int)x0f, y0 = (int)y0f;
        int x1 = min(x0 + 1, WW - 1);
        int y1 = min(y0 + 1, HH - 1);
        int r0 = y0 << 10, r1 = y1 << 10;

        float omx = 1.0f - wx, omy = 1.0f - wy;

        float a00 = base0[r0 + x0], a01 = base0[r0 + x1];
        float a10 = base0[r1 + x0], a11 = base0[r1 + x1];
        float c0 = (a00 * omx + a01 * wx) * omy + (a10 * omx + a11 * wx) * wy;

        float b00 = base1[r0 + x0], b01 = base1[r0 + x1];
        float b10 = base1[r1 + x0], b11 = base1[r1 + x1];
        float c1 = (b00 * omx + b01 * wx) * omy + (b10 * omx + b11 * wx) * wy;

        pout[(size_t)(b * 2 + 0) * PLANE + (size_t)i * WW + j] = c0;
        pout[(size_t)(b * 2 + 1) * PLANE + (size_t)i * WW + j] = c1;
    }
}

extern "C" void kernel_launch(void* const* d_in, const int* in_sizes, int n_in,
                              void* d_out, int out_size, void* d_ws, size_t ws_size,
                              hipStream_t stream) {
    // inputs: d_in[0]=identity (unused, analytic), d_in[1]=u  (both f32, B*C*H*W)
    const float* u = (const float*)d_in[1];
    float* A = (float*)d_out;  // ping
    float* Bb = (float*)d_ws;  // pong (needs 32 MB of workspace)

    const int nvec4 = (NB * 2 * PLANE) / 4;
    init_phi<<<nvec4 / 256, 256, 0, stream>>>(u, A);

    dim3 grid(WW / 32, HH / 32, NB);
    dim3 blk(32, 8, 1);
    float* pin = A;
    float* pout = Bb;
    for (int k = 0; k < 6; ++k) {  // 6 squarings; parity lands result in d_out
        compose<<<grid, blk, 0, stream>>>(pin, pout);
        float* t = pin; pin = pout; pout = t;
    }
    // pin == A == d_out here.
}